// STGCNBlock_52974126628953
// MI455X (gfx1250) — compile-verified
//
#include <hip/hip_runtime.h>
#include <stdint.h>

typedef float v2f __attribute__((ext_vector_type(2)));
typedef float v8f __attribute__((ext_vector_type(8)));

#define EPS 1e-5f

__device__ __forceinline__ v8f wmma4(v2f a, v2f b, v8f c) {
  // D = A(16x4,f32) * B(4x16,f32) + C(16x16,f32)
  return __builtin_amdgcn_wmma_f32_16x16x4_f32(false, a, false, b, (short)0, c,
                                               false, false);
}

// Stage one (n,t) activation slice (64 channels x 25 joints) into LDS as
// dst[ci][v] using CDNA5 async global->LDS loads (ASYNCcnt path, no VGPR
// round-trip).  Rows are 25 contiguous floats: 6 x B128 + 1 x B32 = exact.
__device__ __forceinline__ void async_stage_slice(const float* __restrict__ src,
                                                  float (*dst)[32], int n,
                                                  int t, int tid) {
  for (int pass = 0; pass < 4; ++pass) {
    const int item = tid + pass * 128;  // 64 rows * 7 chunks = 448 items
    if (item < 448) {
      const int ci = item / 7;
      const int c = item % 7;
      const int v0 = (c < 6) ? 4 * c : 24;
      const unsigned long long ga = (unsigned long long)(uintptr_t)(
          src + (((size_t)n * 64 + ci) * 256 + t) * 25 + v0);
      const unsigned la = (unsigned)(uintptr_t)&dst[ci][v0];
      if (c < 6)
        asm volatile("global_load_async_to_lds_b128 %0, %1, off" ::"v"(la),
                     "v"(ga)
                     : "memory");
      else
        asm volatile("global_load_async_to_lds_b32 %0, %1, off" ::"v"(la),
                     "v"(ga)
                     : "memory");
    }
  }
}

__device__ __forceinline__ void wait_async() {
  asm volatile("s_wait_asynccnt 0" ::: "memory");
}

// ---------------------------------------------------------------------------
// Kernel 1: fused UnitGCN.  One block per (n,t).
//   GEMM1 (WMMA): y[co][v] = sum_ci gcn_w[co][ci] * x[ci][v]   (co = 192)
//   GEMM2 (WMMA): p[c][w]  = sum_{kv} y[k*64+c][v] * A[k][v][w]  (kv=75 pad 80)
//   g = relu(BN(p + bias(x)A) + x)
// ---------------------------------------------------------------------------
__global__ __launch_bounds__(128) void gcn_kernel(
    const float* __restrict__ x, const float* __restrict__ A,
    const float* __restrict__ gw, const float* __restrict__ gb,
    const float* __restrict__ gbn, float* __restrict__ gout) {
  __shared__ float xs2[64][32];  // [ci][v], cols 25..31 never read for stores
  __shared__ float ys[192][32];  // [co][v]
  __shared__ float colA[3][32];  // column sums of A over v

  const int bid = blockIdx.x;
  const int n = bid / 256, t = bid % 256;
  const int tid = threadIdx.x;
  const int lane = tid & 31;
  const int wave = tid >> 5;
  const int lm = lane & 15;
  const int hi = lane >> 4;

  async_stage_slice(x, xs2, n, t, tid);
  for (int i = tid; i < 3 * 32; i += 128) {
    int k = i >> 5, w = i & 31;
    float s = 0.f;
    if (w < 25)
      for (int v = 0; v < 25; ++v) s += A[(k * 25 + v) * 25 + w];
    colA[k][w] = s;
  }
  wait_async();
  __syncthreads();

  // GEMM1: A = gcn_w rows (global, row-major), B = xs2 (LDS).  M=co, N=v.
  for (int nt = 0; nt < 3; ++nt) {
    const int co0 = (wave * 3 + nt) * 16;
    v8f acc0 = {}, acc1 = {};
    for (int ks = 0; ks < 16; ++ks) {
      const int k0 = ks * 4 + hi * 2;
      v2f af, b0v, b1v;
      const float* wr = gw + (size_t)(co0 + lm) * 64 + k0;
      af.x = wr[0];
      af.y = wr[1];
      b0v.x = xs2[k0][lm];
      b0v.y = xs2[k0 + 1][lm];
      b1v.x = xs2[k0][16 + lm];
      b1v.y = xs2[k0 + 1][16 + lm];
      acc0 = wmma4(af, b0v, acc0);
      acc1 = wmma4(af, b1v, acc1);
    }
#pragma unroll
    for (int r = 0; r < 8; ++r) {
      const int co = co0 + r + hi * 8;  // D: M = r + 8*hi
      ys[co][lm] = acc0[r];             // N = v
      ys[co][16 + lm] = acc1[r];
    }
  }
  __syncthreads();

  // GEMM2: contract over kv = k*25+v (75, padded to 80). M = c, N = w.
  {
    const int c0 = wave * 16;
    v8f acc0 = {}, acc1 = {};
    for (int ks = 0; ks < 20; ++ks) {
      const int kva = ks * 4 + hi * 2;
      const int kvb = kva + 1;
      v2f af, bA, bB;
      af.x = (kva < 75) ? ys[(kva / 25) * 64 + c0 + lm][kva % 25] : 0.f;
      af.y = (kvb < 75) ? ys[(kvb / 25) * 64 + c0 + lm][kvb % 25] : 0.f;
      bA.x = (kva < 75) ? A[kva * 25 + lm] : 0.f;  // A flat: (k*25+v)*25+w
      bA.y = (kvb < 75) ? A[kvb * 25 + lm] : 0.f;
      const int w1 = 16 + lm;
      bB.x = (kva < 75 && w1 < 25) ? A[kva * 25 + w1] : 0.f;
      bB.y = (kvb < 75 && w1 < 25) ? A[kvb * 25 + w1] : 0.f;
      acc0 = wmma4(af, bA, acc0);
      acc1 = wmma4(af, bB, acc1);
    }
#pragma unroll
    for (int r = 0; r < 8; ++r) {
      const int c = c0 + r + hi * 8;
      const float s = gbn[c] * rsqrtf(gbn[192 + c] + EPS);
      const float off = gbn[64 + c] - gbn[128 + c] * s;
      const float b0 = gb[c], b1 = gb[64 + c], b2 = gb[128 + c];
      {
        const int w = lm;
        float val =
            acc0[r] + b0 * colA[0][w] + b1 * colA[1][w] + b2 * colA[2][w];
        val = val * s + off + xs2[c][w];
        gout[(((size_t)n * 64 + c) * 256 + t) * 25 + w] = fmaxf(val, 0.f);
      }
      {
        const int w = 16 + lm;
        if (w < 25) {
          float val =
              acc1[r] + b0 * colA[0][w] + b1 * colA[1][w] + b2 * colA[2][w];
          val = val * s + off + xs2[c][w];
          gout[(((size_t)n * 64 + c) * 256 + t) * 25 + w] = fmaxf(val, 0.f);
        }
      }
    }
  }
}

// ---------------------------------------------------------------------------
// Kernel 2: concatenated branch 1x1 convs (64->64) with fused BN(+bias)/ReLU.
// Channel layout: [b0:14][b1:10][b2:10][b3:10][b4:10][b5:10]
// ---------------------------------------------------------------------------
__global__ __launch_bounds__(128) void branch1x1_kernel(
    const float* __restrict__ g, const float* __restrict__ w0,
    const float* __restrict__ w1, const float* __restrict__ w2,
    const float* __restrict__ w3, const float* __restrict__ w4,
    const float* __restrict__ w5, const float* __restrict__ bb0,
    const float* __restrict__ bb1, const float* __restrict__ bb2,
    const float* __restrict__ bb3, const float* __restrict__ bb4,
    const float* __restrict__ bb5, const float* __restrict__ bn0,
    const float* __restrict__ bn1, const float* __restrict__ bn2,
    const float* __restrict__ bn3, const float* __restrict__ bn4,
    float* __restrict__ h) {
  __shared__ float gs2[64][32];   // [ci][v] async-staged
  __shared__ float wcat[64][64];  // [co][ci] natural row-major
  __shared__ float s2[64], o2[64];

  const int bid = blockIdx.x;
  const int n = bid / 256, t = bid % 256;
  const int tid = threadIdx.x;
  const int lane = tid & 31;
  const int wave = tid >> 5;
  const int lm = lane & 15;
  const int hi = lane >> 4;

  const float* wp[6] = {w0, w1, w2, w3, w4, w5};
  const float* bp[6] = {bb0, bb1, bb2, bb3, bb4, bb5};
  const float* bnp[5] = {bn0, bn1, bn2, bn3, bn4};

  async_stage_slice(g, gs2, n, t, tid);
  for (int i = tid; i < 64 * 64; i += 128) {
    int co = i >> 6, ci = i & 63;
    int br = (co < 14)   ? 0
             : (co < 24) ? 1
             : (co < 34) ? 2
             : (co < 44) ? 3
             : (co < 54) ? 4
                         : 5;
    int lc = co - ((br == 0) ? 0 : (4 + br * 10));
    wcat[co][ci] = wp[br][lc * 64 + ci];
  }
  if (tid < 64) {
    int co = tid;
    int br = (co < 14)   ? 0
             : (co < 24) ? 1
             : (co < 34) ? 2
             : (co < 44) ? 3
             : (co < 54) ? 4
                         : 5;
    int lc = co - ((br == 0) ? 0 : (4 + br * 10));
    float bias = bp[br][lc];
    float s, o;
    if (br < 5) {
      int cb = (br == 0) ? 14 : 10;
      const float* bn = bnp[br];
      float sc = bn[lc] * rsqrtf(bn[3 * cb + lc] + EPS);
      s = sc;
      o = (bn[cb + lc] - bn[2 * cb + lc] * sc) + bias * sc;
    } else {
      s = 1.f;
      o = bias;
    }
    s2[co] = s;
    o2[co] = o;
  }
  wait_async();
  __syncthreads();

  {
    const int co0 = wave * 16;  // M = co tile, N = v tiles
    v8f acc0 = {}, acc1 = {};
    for (int ks = 0; ks < 16; ++ks) {
      const int k0 = ks * 4 + hi * 2;
      v2f af, b0v, b1v;
      af.x = wcat[co0 + lm][k0];
      af.y = wcat[co0 + lm][k0 + 1];
      b0v.x = gs2[k0][lm];
      b0v.y = gs2[k0 + 1][lm];
      b1v.x = gs2[k0][16 + lm];
      b1v.y = gs2[k0 + 1][16 + lm];
      acc0 = wmma4(af, b0v, acc0);
      acc1 = wmma4(af, b1v, acc1);
    }
#pragma unroll
    for (int r = 0; r < 8; ++r) {
      const int co = co0 + r + hi * 8;
      const float s = s2[co], o = o2[co];
      {
        const int v = lm;
        float val = acc0[r] * s + o;
        if (co < 54) val = fmaxf(val, 0.f);
        h[(((size_t)n * 64 + co) * 256 + t) * 25 + v] = val;
      }
      {
        const int v = 16 + lm;
        if (v < 25) {
          float val = acc1[r] * s + o;
          if (co < 54) val = fmaxf(val, 0.f);
          h[(((size_t)n * 64 + co) * 256 + t) * 25 + v] = val;
        }
      }
    }
  }
}

// ---------------------------------------------------------------------------
// Kernel 3: feat (temporal convs / maxpool / passthrough) + tr-BN/ReLU ->
// transform GEMM (WMMA) -> out-BN -> +x residual -> ReLU -> d_out.
// ---------------------------------------------------------------------------
__global__ __launch_bounds__(128) void mstcn_out_kernel(
    const float* __restrict__ h, const float* __restrict__ x,
    const float* __restrict__ wt0, const float* __restrict__ wt1,
    const float* __restrict__ wt2, const float* __restrict__ wt3,
    const float* __restrict__ bt0, const float* __restrict__ bt1,
    const float* __restrict__ bt2, const float* __restrict__ bt3,
    const float* __restrict__ trbn, const float* __restrict__ trw,
    const float* __restrict__ trb, const float* __restrict__ obn,
    float* __restrict__ out) {
  __shared__ float feats[64][32];  // [ci][v], BN+ReLU applied; pad cols zeroed
  __shared__ float strs[64], otrs[64];

  const int bid = blockIdx.x;
  const int n = bid / 256, t = bid % 256;
  const int tid = threadIdx.x;
  const int lane = tid & 31;
  const int wave = tid >> 5;
  const int lm = lane & 15;
  const int hi = lane >> 4;

  const float* wtp[4] = {wt0, wt1, wt2, wt3};
  const float* btp[4] = {bt0, bt1, bt2, bt3};

  if (tid < 64) {
    float s = trbn[tid] * rsqrtf(trbn[192 + tid] + EPS);
    strs[tid] = s;
    otrs[tid] = trbn[64 + tid] - trbn[128 + tid] * s;
  }
  __syncthreads();

  for (int idx = tid; idx < 64 * 32; idx += 128) {
    const int ci = idx >> 5, v = idx & 31;
    float f = 0.f;
    if (v < 25) {
      if (ci < 44) {  // dilated 3-tap temporal conv branches
        const int br = (ci < 14) ? 0 : (ci < 24) ? 1 : (ci < 34) ? 2 : 3;
        const int cb = (br == 0) ? 14 : 10;
        const int off = (br == 0) ? 0 : (4 + br * 10);
        const int lc = ci - off;
        const int d = br + 1;
        const float* wt = wtp[br];
        float acc = btp[br][lc];
        for (int cc = 0; cc < cb; ++cc) {
          const float* hb = h + (((size_t)n * 64 + off + cc) * 256) * 25 + v;
          const float* wr = wt + (lc * cb + cc) * 3;
#pragma unroll
          for (int tap = 0; tap < 3; ++tap) {
            const int tt = t + (tap - 1) * d;
            const float hv = (tt >= 0 && tt < 256) ? hb[tt * 25] : 0.f;
            acc = fmaf(hv, wr[tap], acc);
          }
        }
        f = acc;
      } else if (ci < 54) {  // maxpool3 branch (-inf padding == skip OOB)
        const float* hb = h + (((size_t)n * 64 + ci) * 256) * 25 + v;
        float m = hb[t * 25];
        if (t > 0) m = fmaxf(m, hb[(t - 1) * 25]);
        if (t < 255) m = fmaxf(m, hb[(t + 1) * 25]);
        f = m;
      } else {  // 1x1 branch passthrough
        f = h[(((size_t)n * 64 + ci) * 256 + t) * 25 + v];
      }
      f = fmaxf(f * strs[ci] + otrs[ci], 0.f);  // tr BN + ReLU
    }
    feats[ci][v] = f;
  }
  __syncthreads();

  {
    const int co0 = wave * 16;  // M = co tile, N = v tiles
    v8f acc0 = {}, acc1 = {};
    for (int ks = 0; ks < 16; ++ks) {
      const int k0 = ks * 4 + hi * 2;
      v2f af, b0v, b1v;
      const float* wr = trw + (size_t)(co0 + lm) * 64 + k0;
      af.x = wr[0];
      af.y = wr[1];
      b0v.x = feats[k0][lm];
      b0v.y = feats[k0 + 1][lm];
      b1v.x = feats[k0][16 + lm];
      b1v.y = feats[k0 + 1][16 + lm];
      acc0 = wmma4(af, b0v, acc0);
      acc1 = wmma4(af, b1v, acc1);
    }
#pragma unroll
    for (int r = 0; r < 8; ++r) {
      const int co = co0 + r + hi * 8;
      const float s = obn[co] * rsqrtf(obn[192 + co] + EPS);
      const float o = obn[64 + co] - obn[128 + co] * s;
      const float tb = trb[co];
      {
        const int v = lm;
        float val = (acc0[r] + tb) * s + o;
        val += x[(((size_t)n * 64 + co) * 256 + t) * 25 + v];
        out[(((size_t)n * 64 + co) * 256 + t) * 25 + v] = fmaxf(val, 0.f);
      }
      {
        const int v = 16 + lm;
        if (v < 25) {
          float val = (acc1[r] + tb) * s + o;
          val += x[(((size_t)n * 64 + co) * 256 + t) * 25 + v];
          out[(((size_t)n * 64 + co) * 256 + t) * 25 + v] = fmaxf(val, 0.f);
        }
      }
    }
  }
}

extern "C" void kernel_launch(void* const* d_in, const int* in_sizes, int n_in,
                              void* d_out, int out_size, void* d_ws,
                              size_t ws_size, hipStream_t stream) {
  (void)in_sizes;
  (void)n_in;
  (void)out_size;
  (void)ws_size;
  const float* x = (const float*)d_in[0];
  const float* A = (const float*)d_in[1];
  const float* gw = (const float*)d_in[2];
  const float* gb = (const float*)d_in[3];
  const float* gbn = (const float*)d_in[4];
  const float* trbn = (const float*)d_in[5];
  const float* trw = (const float*)d_in[6];
  const float* trb = (const float*)d_in[7];
  const float* obn = (const float*)d_in[8];
  const float* bw1[6];
  const float* bb1[6];
  const float* bbn[5];
  const float* bwt[4];
  const float* bbt[4];
  int p = 9;
  for (int i = 0; i < 4; ++i) {
    bw1[i] = (const float*)d_in[p++];
    bb1[i] = (const float*)d_in[p++];
    bbn[i] = (const float*)d_in[p++];
    bwt[i] = (const float*)d_in[p++];
    bbt[i] = (const float*)d_in[p++];
  }
  bw1[4] = (const float*)d_in[p++];
  bb1[4] = (const float*)d_in[p++];
  bbn[4] = (const float*)d_in[p++];
  bw1[5] = (const float*)d_in[p++];
  bb1[5] = (const float*)d_in[p++];

  float* g = (float*)d_ws;                    // (N,64,T,V) = 52.4 MB
  float* h = g + (size_t)32 * 64 * 256 * 25;  // (N,64,T,V) = 52.4 MB

  dim3 grid(32 * 256), block(128);
  gcn_kernel<<<grid, block, 0, stream>>>(x, A, gw, gb, gbn, g);
  branch1x1_kernel<<<grid, block, 0, stream>>>(
      g, bw1[0], bw1[1], bw1[2], bw1[3], bw1[4], bw1[5], bb1[0], bb1[1],
      bb1[2], bb1[3], bb1[4], bb1[5], bbn[0], bbn[1], bbn[2], bbn[3], bbn[4],
      h);
  mstcn_out_kernel<<<grid, block, 0, stream>>>(
      h, x, bwt[0], bwt[1], bwt[2], bwt[3], bbt[0], bbt[1], bbt[2], bbt[3],
      trbn, trw, trb, obn, (float*)d_out);
}